// LightGCN_27066883900223
// MI455X (gfx1250) — compile-verified
//
#include <hip/hip_runtime.h>
#include <math.h>

#define NU 40000
#define NI 20000
#define NN (NU + NI)
#define DD 128
#define NE 320000
#define NLAYERS 3
#define NBATCH 4096
#define FEPS 1e-7f

typedef __attribute__((ext_vector_type(2))) float v2f;
typedef __attribute__((ext_vector_type(8))) float v8f;

// ---- degree / normalization -------------------------------------------------
__global__ void k_init_deg(float* __restrict__ deg, float* __restrict__ lacc) {
    int n = blockIdx.x * 256 + threadIdx.x;
    if (n < NN) deg[n] = FEPS;
    if (n == 0) lacc[0] = 0.0f;
}

__global__ void k_count_deg(const int* __restrict__ au, const int* __restrict__ ai,
                            float* __restrict__ deg) {
    int e = blockIdx.x * 256 + threadIdx.x;  // grid is exact: NE/256
    atomicAdd(deg + au[e], 1.0f);
    atomicAdd(deg + (NU + ai[e]), 1.0f);
}

__global__ void k_dinv(float* __restrict__ deg) {
    int n = blockIdx.x * 256 + threadIdx.x;
    if (n < NN) deg[n] = 1.0f / sqrtf(deg[n]);   // deg^-0.5, full precision
}

// ---- x = concat(user_emb, item_emb); acc = x --------------------------------
__global__ void k_init_x(const float* __restrict__ ue, const float* __restrict__ ie,
                         float* __restrict__ x, float* __restrict__ acc) {
    int i = blockIdx.x * 256 + threadIdx.x;      // grid exact: NN*DD/256
    float v = (i < NU * DD) ? ue[i] : ie[i - NU * DD];
    x[i] = v;
    acc[i] = v;
}

__global__ void k_zero(float* __restrict__ p) {
    p[blockIdx.x * 256 + threadIdx.x] = 0.0f;
}

// ---- SpMM: one wave per edge, both directions; gathers hit L2 (92MB < 192MB)
__global__ void k_spmm(const int* __restrict__ au, const int* __restrict__ ai,
                       const float* __restrict__ dinv,
                       const float* __restrict__ x, float* __restrict__ xn) {
    int lane = threadIdx.x & 31;
    int e = blockIdx.x * 8 + (threadIdx.x >> 5); // grid exact: NE/8 blocks
    int u = au[e];
    int it = NU + ai[e];
    float v = dinv[u] * dinv[it];
    const float4* xu = (const float4*)(x + (size_t)u * DD);
    const float4* xi = (const float4*)(x + (size_t)it * DD);
    float4 a = xi[lane];   // 32 lanes x 4 = 128 floats
    float4 b = xu[lane];
    float* ou = xn + (size_t)u * DD + lane * 4;
    float* oi = xn + (size_t)it * DD + lane * 4;
    atomicAdd(ou + 0, v * a.x); atomicAdd(ou + 1, v * a.y);
    atomicAdd(ou + 2, v * a.z); atomicAdd(ou + 3, v * a.w);
    atomicAdd(oi + 0, v * b.x); atomicAdd(oi + 1, v * b.y);
    atomicAdd(oi + 2, v * b.z); atomicAdd(oi + 3, v * b.w);
}

__global__ void k_accum(const float* __restrict__ xn, float* __restrict__ acc) {
    int i = blockIdx.x * 256 + threadIdx.x;
    acc[i] += xn[i];
}

// ---- scoring: 16 batch rows per wave via V_WMMA_F32_16X16X4_F32 -------------
// A(16x4) = tu chunk, B(4x16) = (pi-ni)^T chunk, diag(D) = per-row dot products.
// A layout: lane l<16 -> M=l, K={k0,k0+1}; lanes 16-31 -> M=l-16, K={k0+2,k0+3}.
// B layout mirrors with lane=N, so both use the same float2 gather.
__global__ void k_score(const float* __restrict__ acc,
                        const int* __restrict__ user, const int* __restrict__ pos,
                        const int* __restrict__ neg, float* __restrict__ lacc) {
    int lane = threadIdx.x & 31;
    int wave = threadIdx.x >> 5;
    int group = blockIdx.x * 8 + wave;           // 256 groups of 16 rows
    int lm = lane & 15;
    int kofs = (lane >> 4) * 2;
    int b = group * 16 + lm;
    const float* tu = acc + (size_t)user[b] * DD;
    const float* pr = acc + (size_t)(NU + pos[b] - 1) * DD;
    const float* nr = acc + (size_t)(NU + neg[b] - 1) * DD;
    v8f c = {0.f, 0.f, 0.f, 0.f, 0.f, 0.f, 0.f, 0.f};
#pragma unroll
    for (int k0 = 0; k0 < DD; k0 += 4) {
        int k = k0 + kofs;
        v2f a  = *(const v2f*)(tu + k);
        v2f bp = *(const v2f*)(pr + k);
        v2f bn = *(const v2f*)(nr + k);
        v2f bb = bp - bn;
        c = __builtin_amdgcn_wmma_f32_16x16x4_f32(
                false, a, false, bb, (short)0, c, false, false);
    }
    // Extract diagonal: m<8 at lane m (c[m]); m>=8 at lane m+16 (c[m-8]).
    bool act = (lane < 8) || (lane >= 24);
    int ci = lane & 7;
    float s = c[0];
#pragma unroll
    for (int i = 1; i < 8; ++i) if (ci == i) s = c[i];
    if (act) {
        s *= 0.0625f;  // fold (1/(LAYERS+1))^2 = (1/4)^2
        // softplus(-s), numerically stable:  -log_sigmoid(s)
        float l = fmaxf(-s, 0.0f) + log1pf(expf(-fabsf(s)));
        atomicAdd(lacc, l);
    }
}

__global__ void k_final(const float* __restrict__ lacc, float* __restrict__ out) {
    out[0] = lacc[0] * (1.0f / (float)NBATCH);
}

// -----------------------------------------------------------------------------
extern "C" void kernel_launch(void* const* d_in, const int* in_sizes, int n_in,
                              void* d_out, int out_size, void* d_ws, size_t ws_size,
                              hipStream_t stream) {
    const float* ue  = (const float*)d_in[0];
    const float* ie  = (const float*)d_in[1];
    const int* au    = (const int*)d_in[2];
    const int* ai    = (const int*)d_in[3];
    const int* user  = (const int*)d_in[4];
    const int* pos   = (const int*)d_in[5];
    const int* neg   = (const int*)d_in[6];
    float* out = (float*)d_out;

    // Workspace carve (floats): deg/dinv, xA, xB, acc, loss accumulator.
    float* ws = (float*)d_ws;
    size_t off = 0;
    float* deg = ws + off;  off += 60032;                 // 60000 rounded up
    float* xA  = ws + off;  off += (size_t)NN * DD;
    float* xB  = ws + off;  off += (size_t)NN * DD;
    float* acc = ws + off;  off += (size_t)NN * DD;
    float* lacc = ws + off; off += 64;

    const int ND_BLK = (NN * DD) / 256;  // 30000, exact

    k_init_deg<<<(NN + 255) / 256, 256, 0, stream>>>(deg, lacc);
    k_count_deg<<<NE / 256, 256, 0, stream>>>(au, ai, deg);
    k_dinv<<<(NN + 255) / 256, 256, 0, stream>>>(deg);
    k_init_x<<<ND_BLK, 256, 0, stream>>>(ue, ie, xA, acc);

    float* x = xA;
    float* xn = xB;
    for (int l = 0; l < NLAYERS; ++l) {
        k_zero<<<ND_BLK, 256, 0, stream>>>(xn);
        k_spmm<<<NE / 8, 256, 0, stream>>>(au, ai, deg, x, xn);
        k_accum<<<ND_BLK, 256, 0, stream>>>(xn, acc);
        float* t = x; x = xn; xn = t;
    }

    k_score<<<NBATCH / 128, 256, 0, stream>>>(acc, user, pos, neg, lacc);
    k_final<<<1, 1, 0, stream>>>(lacc, out);
}